// EMD_46712064311444
// MI455X (gfx1250) — compile-verified
//
#include <hip/hip_runtime.h>
#include <hip/hip_bf16.h>

typedef __attribute__((ext_vector_type(2))) float v2f;
typedef __attribute__((ext_vector_type(8))) float v8f;

#define WAVES    8
#define N_ITERS  100
#define EPS_F    1e-16f
#define REG_F    10.0f
#define REG_INV  0.1f

__device__ __forceinline__ float bcast(float x, int lane) {
  return __int_as_float(__builtin_amdgcn_readlane(__float_as_int(x), lane));
}

__device__ __forceinline__ float wave_sum32(float x) {
#pragma unroll
  for (int off = 16; off > 0; off >>= 1) x += __shfl_xor(x, off, 32);
  return x;
}

__global__ void __launch_bounds__(32 * WAVES)
emd_sinkhorn_kernel(const float* __restrict__ pred,
                    const float* __restrict__ score,
                    const float* __restrict__ tgt,
                    float* __restrict__ ws,
                    float* __restrict__ out,
                    int B, int use_atomic) {
  __shared__ float Klds[WAVES][32 * 33];

  const int tid    = threadIdx.x;
  const int wv     = tid >> 5;
  const int lane   = tid & 31;
  const int lanelo = lane & 15;
  const int hi     = lane >> 4;

  int b = blockIdx.x * WAVES + wv;
  const int valid = (b < B);
  if (b >= B) b = B - 1;

  const float* pr = pred + (size_t)b * 32 * 64;
  const float* tg = tgt  + (size_t)b * 32 * 64;

  // ---- Phase 1: Gram matrix via WMMA f32 16x16x4, fused norm accumulation ----
  v8f acc00 = {}, acc01 = {}, acc10 = {}, acc11 = {};
  float np0 = 0.f, np1 = 0.f;   // |pred row|^2 partials
  float tn0 = 0.f, tn1 = 0.f;   // |tgt row|^2 partials
  float ab0 = 0.f, ab1 = 0.f;   // sum|tgt row| partials

#pragma unroll
  for (int kk = 0; kk < 16; ++kk) {
    const int ko = kk * 4 + 2 * hi;
    v2f a0 = *(const v2f*)(pr + (size_t)lanelo * 64 + ko);
    v2f a1 = *(const v2f*)(pr + (size_t)(lanelo + 16) * 64 + ko);
    v2f b0 = *(const v2f*)(tg + (size_t)lanelo * 64 + ko);
    v2f b1 = *(const v2f*)(tg + (size_t)(lanelo + 16) * 64 + ko);

    acc00 = __builtin_amdgcn_wmma_f32_16x16x4_f32(false, a0, false, b0, (short)0, acc00, false, false);
    acc01 = __builtin_amdgcn_wmma_f32_16x16x4_f32(false, a0, false, b1, (short)0, acc01, false, false);
    acc10 = __builtin_amdgcn_wmma_f32_16x16x4_f32(false, a1, false, b0, (short)0, acc10, false, false);
    acc11 = __builtin_amdgcn_wmma_f32_16x16x4_f32(false, a1, false, b1, (short)0, acc11, false, false);

    np0 += a0.x * a0.x + a0.y * a0.y;
    np1 += a1.x * a1.x + a1.y * a1.y;
    tn0 += b0.x * b0.x + b0.y * b0.y;
    tn1 += b1.x * b1.x + b1.y * b1.y;
    ab0 += fabsf(b0.x) + fabsf(b0.y);
    ab1 += fabsf(b1.x) + fabsf(b1.y);
  }
  np0 += __shfl_xor(np0, 16, 32);
  np1 += __shfl_xor(np1, 16, 32);
  tn0 += __shfl_xor(tn0, 16, 32);
  tn1 += __shfl_xor(tn1, 16, 32);
  ab0 += __shfl_xor(ab0, 16, 32);
  ab1 += __shfl_xor(ab1, 16, 32);

  const float pn  = hi ? np1 : np0;   // per-lane: |pred_lane|^2
  const float abv = hi ? ab1 : ab0;   // per-lane: sum|tgt_lane|

  // ---- Phase 2: cost -> K = exp(-cost/REG), scatter to LDS (pitch 33) ----
  float* kw = &Klds[wv][0];
  {
    v8f accs[4] = {acc00, acc01, acc10, acc11};
#pragma unroll
    for (int t = 0; t < 4; ++t) {
      const int   tr  = t >> 1, tc = t & 1;
      const float tnv = tc ? tn1 : tn0;   // per-lane: |tgt_{tc*16+lanelo}|^2
#pragma unroll
      for (int j = 0; j < 8; ++j) {
        const float pl = bcast(pn, tr * 16 + j);
        const float ph = bcast(pn, tr * 16 + j + 8);
        const float pv = hi ? ph : pl;
        const float g  = accs[t][j];
        const float d2 = fmaxf(pv + tnv - 2.0f * g, EPS_F);
        const float c  = sqrtf(d2);
        const float kv = __expf(-c * REG_INV);
        const int n = tr * 16 + j + 8 * hi;
        const int m = tc * 16 + lanelo;
        kw[n * 33 + m] = kv;
      }
    }
  }
  __syncthreads();

  // ---- Phase 3: pull K into both per-lane layouts (registers) ----
  float kr[32], kc[32];
#pragma unroll
  for (int i = 0; i < 32; ++i) {
    kr[i] = kw[lane * 33 + i];   // row layout:    lane n holds K[n][i]
    kc[i] = kw[i * 33 + lane];   // column layout: lane m holds K[i][m]
  }

  // ---- Marginals p, q and validity mask ----
  const float sc   = score[(size_t)b * 32 + lane];
  const float ssum = wave_sum32(sc);
  const float p    = sc / ssum;
  const float nz   = (abv > 0.0f) ? 1.0f : 0.0f;
  const float qsum = wave_sum32(nz);
  const float q    = nz / qsum;
  const int maskOK = (ssum > 0.0f) && (qsum > 0.0f);

  // ---- Phase 4: 100 Sinkhorn iterations, all in registers ----
  float u = 1.0f / 32.0f;
  float v = 0.0f;
  for (int it = 0; it < N_ITERS; ++it) {
    float s = 0.0f;
#pragma unroll
    for (int n = 0; n < 32; ++n) s = fmaf(kc[n], bcast(u, n), s);   // (K^T u)[lane]
    v = q / (s + EPS_F);
    float t = 0.0f;
#pragma unroll
    for (int m = 0; m < 32; ++m) t = fmaf(kr[m], bcast(v, m), t);   // (K v)[lane]
    u = p / (t + EPS_F);
  }
  {
    float s = 0.0f;
#pragma unroll
    for (int n = 0; n < 32; ++n) s = fmaf(kc[n], bcast(u, n), s);
    v = q / (s + EPS_F);
  }

  // ---- Phase 5: loss = sum_n u_n * sum_m K_nm * cost_nm * v_m ;  cost = -REG*ln(K) ----
  float w = 0.0f;
#pragma unroll
  for (int m = 0; m < 32; ++m) {
    const float km = kr[m];
    const float cm = -REG_F * __logf(km);
    w = fmaf(km * cm, bcast(v, m), w);
  }
  const float tot = wave_sum32(u * w);

  if (lane == 0 && valid) {
    const float val = maskOK ? tot : 0.0f;
    if (use_atomic) atomicAdd(out, val / (float)B);
    else            ws[b] = val;
  }
}

__global__ void __launch_bounds__(256)
emd_reduce_kernel(const float* __restrict__ ws, float* __restrict__ out, int B) {
  __shared__ float sm[256];
  float s = 0.0f;
  for (int i = threadIdx.x; i < B; i += 256) s += ws[i];
  sm[threadIdx.x] = s;
  __syncthreads();
  for (int st = 128; st > 0; st >>= 1) {
    if (threadIdx.x < st) sm[threadIdx.x] += sm[threadIdx.x + st];
    __syncthreads();
  }
  if (threadIdx.x == 0) out[0] = sm[0] / (float)B;
}

extern "C" void kernel_launch(void* const* d_in, const int* in_sizes, int n_in,
                              void* d_out, int out_size, void* d_ws, size_t ws_size,
                              hipStream_t stream) {
  const float* pred  = (const float*)d_in[0];  // [B,32,64]
  const float* score = (const float*)d_in[1];  // [B,32]
  const float* tgt   = (const float*)d_in[2];  // [B,32,64]
  float* out = (float*)d_out;

  const int B = in_sizes[1] / 32;
  const int blocks = (B + WAVES - 1) / WAVES;

  if (ws_size >= (size_t)B * sizeof(float)) {
    float* ws = (float*)d_ws;
    emd_sinkhorn_kernel<<<blocks, 32 * WAVES, 0, stream>>>(pred, score, tgt, ws, out, B, 0);
    emd_reduce_kernel<<<1, 256, 0, stream>>>(ws, out, B);
  } else {
    hipMemsetAsync(d_out, 0, sizeof(float), stream);
    emd_sinkhorn_kernel<<<blocks, 32 * WAVES, 0, stream>>>(pred, score, tgt, nullptr, out, B, 1);
  }
}